// LayerRGAT_5385888989543
// MI455X (gfx1250) — compile-verified
//
#include <hip/hip_runtime.h>
#include <hip/hip_bf16.h>

// ---------------------------------------------------------------------------
// RGAT layer, algebraically collapsed (segment-softmax weights sum to 1 and
// the message is h[dst], so attention reduces to fixed matrices):
//   Weff     = mean_k(node_att_W) @ W_out[:D] + mean_m(rel_att_W) @ W_out[D:]
//   h_out    = deg>0 ? relu(h @ Weff + b_out) : h
//   new_node = relu(h_out @ W_node + b_node) + h
//   new_edge = relu(he @ W_edge + b_edge) + he
// All GEMMs run on V_WMMA_F32_16X16X4_F32 (f32 tensor path, preserves
// reference numerics). Memory floor ~313 MB -> ~13.5 us @ 23.3 TB/s.
// rev3: W stored TRANSPOSED in LDS (stride 132) so each B fragment is one
// conflict-free ds_load_b64 into an aligned VGPR pair (no reassembly movs),
// plus prefetch distance 2 and dual accumulator chains.
// ---------------------------------------------------------------------------

#define DIMN 128
#define TSTR 132   // transposed-LDS column stride (floats): lane banks tile 0..63

typedef float v2f __attribute__((ext_vector_type(2)));
typedef float v8f __attribute__((ext_vector_type(8)));

__device__ __forceinline__ v2f ldsBT(const float* __restrict__ sWT, int col, int krow) {
  // B fragment (ISA 7.12.2): lane(hi,n) needs (W[krow][col], W[krow+1][col]).
  // Transposed layout makes them consecutive -> single ds_load_b64.
  return *(const v2f*)(sWT + (size_t)col * TSTR + krow);
}

// ---------------------------------------------------------------------------
// Generic fused GEMM: O = epilogue( X[rows,128] @ W[128,128] + bias )
//   DEGSEL=false: O = relu(acc) + aux[row]          (residual)
//   DEGSEL=true : O = deg[row] ? relu(acc) : aux[row]
// 256 threads = 8 wave32; each wave computes a 16x128 strip as 4 passes of
// TWO 16x16 column tiles, 32 K-steps of v_wmma_f32_16x16x4_f32 each, with
// B fragments prefetched TWO K-steps ahead.
// ---------------------------------------------------------------------------
template <bool DEGSEL>
__global__ __launch_bounds__(256) void gemm128_wmma_kernel(
    const float* __restrict__ X,
    const float* __restrict__ Wg,
    const float* __restrict__ bias,
    const float* __restrict__ aux,
    const int*   __restrict__ deg,
    float*       __restrict__ O,
    int rows)
{
  extern __shared__ float smem[];        // 128 cols x TSTR floats (transposed W)
  float* sWT = smem;

  const int tid = threadIdx.x;

  // Cooperative transposed stage: global reads coalesced (consecutive tid ->
  // consecutive floats of a W row); LDS stores strided (one-time cost).
  for (int idx = tid; idx < DIMN * DIMN; idx += 256) {
    const int r = idx >> 7;            // W row  (= K index)
    const int c = idx & 127;           // W col  (= output column)
    sWT[(size_t)c * TSTR + r] = Wg[idx];
  }
  __syncthreads();

  const int wave = tid >> 5;
  const int lane = tid & 31;
  const int hi   = lane >> 4;        // 0/1: which K/M half this lane serves
  const int ln   = lane & 15;        // row (A) / column (B,C,D) within tile
  const int rowBase = blockIdx.x * 128 + wave * 16;
  if (rowBase >= rows) return;       // wave-uniform; rows % 16 == 0 here

  // A fragments: lane(hi,m) holds X[m][4kt+2hi], X[m][4kt+2hi+1]  (ISA 7.12.2)
  v2f afrag[32];
  {
    const float* xrow = X + (size_t)(rowBase + ln) * DIMN + 2 * hi;
#pragma unroll
    for (int kt = 0; kt < 32; ++kt) {
      afrag[kt] = *(const v2f*)(xrow + 4 * kt);
    }
  }

#pragma unroll 1
  for (int p = 0; p < 4; ++p) {          // 4 passes x 2 column tiles
    const int col0 = p * 32 + ln;        // tile A column
    const int col1 = col0 + 16;          // tile B column
    const float bv0 = bias[col0];
    const float bv1 = bias[col1];
    v8f acc0, acc1;
#pragma unroll
    for (int r = 0; r < 8; ++r) { acc0[r] = bv0; acc1[r] = bv1; }

    // Prefetch distance 2: two fragment generations in flight per column tile,
    // two independent WMMA chains -> waits relax to dscnt<=4 and the XDL pipe
    // always has ready work.
    v2f b0[2], b1[2];
    b0[0] = ldsBT(sWT, col0, 2 * hi);
    b1[0] = ldsBT(sWT, col1, 2 * hi);
    b0[1] = ldsBT(sWT, col0, 4 + 2 * hi);
    b1[1] = ldsBT(sWT, col1, 4 + 2 * hi);
#pragma unroll
    for (int kt = 0; kt < 32; ++kt) {
      const int s = kt & 1;
      acc0 = __builtin_amdgcn_wmma_f32_16x16x4_f32(
          false, afrag[kt], false, b0[s], (short)0, acc0, false, false);
      acc1 = __builtin_amdgcn_wmma_f32_16x16x4_f32(
          false, afrag[kt], false, b1[s], (short)0, acc1, false, false);
      if (kt + 2 < 32) {
        const int nk = 4 * (kt + 2) + 2 * hi;
        b0[s] = ldsBT(sWT, col0, nk);
        b1[s] = ldsBT(sWT, col1, nk);
      }
    }

    // Epilogue: D layout -> acc[r] is output row (r + 8*hi).
#pragma unroll
    for (int r = 0; r < 8; ++r) {
      const int row = rowBase + r + 8 * hi;
      const size_t i0 = (size_t)row * DIMN + col0;
      const size_t i1 = (size_t)row * DIMN + col1;
      float v0 = acc0[r]; v0 = v0 > 0.0f ? v0 : 0.0f;   // relu(X@W + b)
      float v1 = acc1[r]; v1 = v1 > 0.0f ? v1 : 0.0f;
      float o0, o1;
      if (DEGSEL) {
        const bool keep = (deg[row] != 0);
        o0 = keep ? v0 : aux[i0];
        o1 = keep ? v1 : aux[i1];
      } else {
        o0 = v0 + aux[i0];
        o1 = v1 + aux[i1];
      }
      O[i0] = o0;
      O[i1] = o1;
    }
  }
}

// ---------------------------------------------------------------------------
// Weff[i][j] = sum_f mean_k(nattW[k][i][f]) * Wout[f][j]
//            + sum_f mean_m(relW[m][i][f])  * Wout[128+f][j]
// Tiny (128x128 out, ~4.2M MACs) — plain VALU kernel.
// ---------------------------------------------------------------------------
__global__ __launch_bounds__(256) void weff_kernel(
    const float* __restrict__ nattW,   // [4,128,128]
    const float* __restrict__ relW,    // [4,128,128]
    const float* __restrict__ Wout,    // [256,128]
    float* __restrict__ Weff)          // [128,128]
{
  const int t = blockIdx.x * 256 + threadIdx.x;
  if (t >= DIMN * DIMN) return;
  const int i = t >> 7;
  const int j = t & 127;
  float acc = 0.0f;
  for (int f = 0; f < DIMN; ++f) {
    float wa = 0.0f, wr = 0.0f;
#pragma unroll
    for (int k = 0; k < 4; ++k) wa += nattW[((size_t)(k * DIMN + i)) * DIMN + f];
#pragma unroll
    for (int m = 0; m < 4; ++m) wr += relW[((size_t)(m * DIMN + i)) * DIMN + f];
    wa *= 0.25f;                       // mean over K=4
    wr *= 0.25f;                       // mean over M=4
    acc += wa * Wout[(size_t)f * DIMN + j] + wr * Wout[(size_t)(DIMN + f) * DIMN + j];
  }
  Weff[t] = acc;
}

__global__ void zero_deg_kernel(int* __restrict__ deg, int n) {
  const int t = blockIdx.x * blockDim.x + threadIdx.x;
  if (t < n) deg[t] = 0;
}

__global__ void mark_deg_kernel(const int* __restrict__ dst,
                                int* __restrict__ deg, int e) {
  const int t = blockIdx.x * blockDim.x + threadIdx.x;
  if (t < e) deg[dst[t]] = 1;          // benign race: every writer stores 1
}

// ---------------------------------------------------------------------------
extern "C" void kernel_launch(void* const* d_in, const int* in_sizes, int n_in,
                              void* d_out, int out_size, void* d_ws, size_t ws_size,
                              hipStream_t stream) {
  (void)n_in; (void)out_size; (void)ws_size;

  const float* h     = (const float*)d_in[0];   // [N,128]
  const float* he    = (const float*)d_in[1];   // [E,128]
  const int*   dst   = (const int*)  d_in[3];   // [E]
  const float* nattW = (const float*)d_in[4];   // [4,128,128]
  const float* relW  = (const float*)d_in[5];   // [4,128,128]
  // d_in[2] (src), d_in[6..9] (edge-gate MLP): provably dead after the
  // segment-softmax collapse (weights sum to 1; message is h[dst]).
  const float* Wout  = (const float*)d_in[10];  // [256,128]
  const float* bout  = (const float*)d_in[11];  // [128]
  const float* Wnode = (const float*)d_in[12];  // [128,128]
  const float* bnode = (const float*)d_in[13];  // [128]
  const float* Wedge = (const float*)d_in[14];  // [128,128]
  const float* bedge = (const float*)d_in[15];  // [128]

  const int N = in_sizes[0] / DIMN;             // 20000
  const int E = in_sizes[1] / DIMN;             // 256000

  float* out_node = (float*)d_out;                           // [N,128]
  float* out_edge = (float*)d_out + (size_t)N * DIMN;        // [E,128]

  float* Weff = (float*)d_ws;                                // 64 KB
  int*   deg  = (int*)((float*)d_ws + DIMN * DIMN);          // N ints

  const size_t shmem = (size_t)DIMN * TSTR * sizeof(float);  // ~66 KB (LDS=320KB/WGP)

  // Prep (tiny)
  weff_kernel<<<(DIMN * DIMN + 255) / 256, 256, 0, stream>>>(nattW, relW, Wout, Weff);
  zero_deg_kernel<<<(N + 255) / 256, 256, 0, stream>>>(deg, N);
  mark_deg_kernel<<<(E + 255) / 256, 256, 0, stream>>>(dst, deg, E);

  // new_edge = relu(he @ W_edge + b_edge) + he          (bulk: 262 MB, 8.4 GFLOP)
  gemm128_wmma_kernel<false><<<(E + 127) / 128, 256, shmem, stream>>>(
      he, Wedge, bedge, /*aux=*/he, /*deg=*/nullptr, out_edge, E);

  // h_out = deg>0 ? relu(h @ Weff + b_out) : h     -> node region of d_out
  gemm128_wmma_kernel<true><<<(N + 127) / 128, 256, shmem, stream>>>(
      h, Weff, bout, /*aux=*/h, deg, out_node, N);

  // new_node = relu(h_out @ W_node + b_node) + h   (in place; each strip reads
  // its own rows into registers before any store -> no hazard)
  gemm128_wmma_kernel<false><<<(N + 127) / 128, 256, shmem, stream>>>(
      out_node, Wnode, bnode, /*aux=*/h, /*deg=*/nullptr, out_node, N);
}